// GraphFullConvolution_43559558316211
// MI455X (gfx1250) — compile-verified
//
#include <hip/hip_runtime.h>

typedef float v2f __attribute__((ext_vector_type(2)));
typedef float v8f __attribute__((ext_vector_type(8)));

namespace {
constexpr int kM    = 65536;
constexpr int kFin  = 32;
constexpr int kFout = 32;
constexpr int kK    = 7;

constexpr int kThreads       = 256;  // 8 wave32 per block
constexpr int kWavesPerBlock = kThreads / 32;
constexpr int kMPerWave      = 4;
constexpr int kMPerBlock     = kWavesPerBlock * kMPerWave;  // 32

// Packed B-fragment LDS layout: [k][c][t][lane] of float2
constexpr int kFragsPerK = 8 * 2 * 32;              // c * t * lane = 512
constexpr int kLdsFloats = kK * kFragsPerK * 2;     // 7168 floats = 28 KB
}  // namespace

// out[n,m,o] = sum_{k,f} x[n, idx[m,k], f] * val[m,k] * w[k,f,o] + bias[o]
// One wave computes the full 16(n) x 32(o) output tile for 4 consecutive m.
__global__ __launch_bounds__(kThreads)
void gfc_wmma_f32_kernel(const float* __restrict__ x,
                         const float* __restrict__ weight,
                         const float* __restrict__ bias,
                         const int*   __restrict__ index_list,
                         const float* __restrict__ index_value,
                         float* __restrict__ out)
{
    // Pre-pack weights into WMMA 16x16x4 B-fragment layout so each fragment
    // is one ds_load_b64 (pair contiguous per lane), no register shuffles.
    //   fragment (k,c,t), lane: { w[k][4c+fh][t*16+o], w[k][4c+fh+1][t*16+o] }
    //   where fh = (lane>>4)*2, o = lane&15.
    __shared__ float ldsb[kLdsFloats];
    for (int p = threadIdx.x; p < kK * kFragsPerK; p += kThreads) {
        const int lane_p = p & 31;
        const int t_p    = (p >> 5) & 1;
        const int c_p    = (p >> 6) & 7;
        const int k_p    = p >> 9;
        const int fh     = (lane_p >> 4) << 1;
        const int o_p    = lane_p & 15;
        const int src = k_p * kFin * kFout + (4 * c_p + fh) * kFout + t_p * 16 + o_p;
        v2f w2;
        w2.x = weight[src];
        w2.y = weight[src + kFout];
        *(v2f*)&ldsb[p * 2] = w2;   // 8B-aligned
    }
    __syncthreads();

    const int lane = threadIdx.x & 31;
    const int wave = threadIdx.x >> 5;
    const int m0   = blockIdx.x * kMPerBlock + wave * kMPerWave;

    // WMMA 16x16x4 f32 fragment geometry (wave32):
    //   A: lane l<16 -> row M=l, K elems {4c, 4c+1}; lane l>=16 -> row M=l-16, K {4c+2, 4c+3}
    //   C/D: VGPR r -> output rows {r (lanes 0-15), r+8 (lanes 16-31)}, col = lane&15
    const int nrow  = lane & 15;          // A-row (= n); also o-within-tile / out col
    const int fhalf = (lane >> 4) << 1;   // 0 for lanes 0-15, 2 for lanes 16-31

    const v8f vzero = {0.f, 0.f, 0.f, 0.f, 0.f, 0.f, 0.f, 0.f};
    v8f acc[kMPerWave][2];
#pragma unroll
    for (int j = 0; j < kMPerWave; ++j)
#pragma unroll
        for (int t = 0; t < 2; ++t)
            acc[j][t] = vzero;

    for (int k = 0; k < kK; ++k) {
        // B fragments for this k: single ds_load_b64 each, shared by all 4 m.
        const float* bk = &ldsb[k * kFragsPerK * 2];
        v2f bfrag[8][2];
#pragma unroll
        for (int c = 0; c < 8; ++c)
#pragma unroll
            for (int t = 0; t < 2; ++t)
                bfrag[c][t] =
                    *(const v2f*)(bk + (((c << 1) + t) << 6) + (lane << 1));

#pragma unroll
        for (int j = 0; j < kMPerWave; ++j) {
            const int   m    = m0 + j;
            const int   gidx = index_list[m * kK + k];     // wave-uniform broadcast load
            const float gval = index_value[m * kK + k];    // wave-uniform broadcast load
            // Gathered x row: 128 contiguous bytes; stays hot in 192MB L2.
            const float* xrow =
                x + ((size_t)nrow * kM + (size_t)gidx) * kFin + fhalf;
#pragma unroll
            for (int c = 0; c < 8; ++c) {
                v2f a = *(const v2f*)(xrow + 4 * c);   // 8B-aligned float2
                a.x *= gval;
                a.y *= gval;
#pragma unroll
                for (int t = 0; t < 2; ++t) {
                    acc[j][t] = __builtin_amdgcn_wmma_f32_16x16x4_f32(
                        /*neg_a=*/false, a, /*neg_b=*/false, bfrag[c][t],
                        /*c_mod=*/(short)0, acc[j][t],
                        /*reuse_a=*/false, /*reuse_b=*/false);
                }
            }
        }
    }

    const float bias0 = bias[nrow];
    const float bias1 = bias[16 + nrow];

#pragma unroll
    for (int j = 0; j < kMPerWave; ++j) {
        const int m = m0 + j;
#pragma unroll
        for (int r = 0; r < 8; ++r) {
            const int n_out = (lane < 16) ? r : (r + 8);   // select, no EXEC divergence
            float* orow = out + ((size_t)n_out * kM + (size_t)m) * kFout + nrow;
            orow[0]  = acc[j][0][r] + bias0;
            orow[16] = acc[j][1][r] + bias1;
        }
    }
}

extern "C" void kernel_launch(void* const* d_in, const int* in_sizes, int n_in,
                              void* d_out, int out_size, void* d_ws, size_t ws_size,
                              hipStream_t stream) {
    (void)in_sizes; (void)n_in; (void)out_size; (void)d_ws; (void)ws_size;
    const float* x           = (const float*)d_in[0];
    const float* weight      = (const float*)d_in[1];
    const float* bias        = (const float*)d_in[2];
    const int*   index_list  = (const int*)d_in[3];
    const float* index_value = (const float*)d_in[4];
    float* out = (float*)d_out;

    dim3 grid(kM / kMPerBlock);   // 65536 / 32 = 2048 blocks, exact
    dim3 block(kThreads);
    gfc_wmma_f32_kernel<<<grid, block, 0, stream>>>(x, weight, bias,
                                                    index_list, index_value, out);
}